// PAttnDecoderRNN_82815559401826
// MI455X (gfx1250) — compile-verified
//
#include <hip/hip_runtime.h>
#include <hip/hip_bf16.h>
#include <math.h>

// Sizes from the reference
#define H 1024
#define E 512
#define V 32000
#define L 512

typedef float v2f __attribute__((ext_vector_type(2)));
typedef float v8f __attribute__((ext_vector_type(8)));

// ---------------------------------------------------------------------------
// WMMA-based GEMV: y[r] = dot(W[r, 0:K], vec) + bias[r]
//   grid.x blocks of 128 threads (4 waves); each block owns 64 rows.
//   K must be a multiple of 64 (all call sites: 512 or 1024).
//   Uses V_WMMA_F32_16X16X4_F32: A = 16x4 tile of W (from LDS),
//   B = vec chunk replicated across all 16 columns -> every column of the
//   16x16 accumulator holds the same 16 dot products.
// ---------------------------------------------------------------------------
__global__ void gemv_wmma_kernel(const float* __restrict__ W,
                                 const float* __restrict__ vec,
                                 const float* __restrict__ bias,
                                 float* __restrict__ y,
                                 int K)
{
    __shared__ float tile[64][68];   // 64x64 fp32 tile, +4 pad floats per row
    __shared__ float xs[64];

    const int tid  = threadIdx.x;    // 0..127
    const int lane = tid & 31;
    const int wave = tid >> 5;       // 0..3
    const int row0 = blockIdx.x * 64;

    const int m    = lane & 15;          // A-matrix M index (ISA 7.12.2)
    const int koff = (lane >> 4) * 2;    // lanes 0-15: K0,K1 ; lanes 16-31: K2,K3

    v8f acc = {};

    for (int kc0 = 0; kc0 < K; kc0 += 64) {
        // Cooperative, coalesced tile load: 1024 float4s over 128 threads.
        #pragma unroll
        for (int j = 0; j < 8; ++j) {
            const int f  = tid + j * 128;   // 0..1023
            const int r  = f >> 4;          // tile row
            const int c4 = f & 15;          // float4 column
            const size_t gaddr = (size_t)(row0 + r) * (size_t)K + (size_t)(kc0 + c4 * 4);
            const float4 w4 = *reinterpret_cast<const float4*>(&W[gaddr]);
            *reinterpret_cast<float4*>(&tile[r][c4 * 4]) = w4;
            if (kc0 + 64 < K) {             // prefetch next K-chunk (global_prefetch_b8)
                __builtin_prefetch(&W[gaddr + 64], 0, 1);
            }
        }
        if (tid < 64) xs[tid] = vec[kc0 + tid];
        __syncthreads();

        const int mrow = wave * 16 + m;
        #pragma unroll
        for (int kc = 0; kc < 64; kc += 4) {
            // A fragment (16x4 f32): lane holds W[mrow][kc+koff], W[mrow][kc+koff+1]
            v2f a = *reinterpret_cast<const v2f*>(&tile[mrow][kc + koff]);
            // B fragment (4x16 f32), replicated across N; K mapping assumed
            // mirror-symmetric to the documented A layout.
            v2f b;
            b.x = xs[kc + koff];
            b.y = xs[kc + koff + 1];
            acc = __builtin_amdgcn_wmma_f32_16x16x4_f32(
                      false, a, false, b, (short)0, acc, false, false);
        }
        __syncthreads();
    }

    // C/D layout: VGPR r -> (M=r, lanes 0-15) and (M=8+r, lanes 16-31);
    // all columns identical, so lanes 0 and 16 carry rows 0-7 / 8-15.
    if ((lane & 15) == 0) {
        const int rbase = row0 + wave * 16 + (lane >> 4) * 8;
        #pragma unroll
        for (int r2 = 0; r2 < 8; ++r2) {
            const float b = bias ? bias[rbase + r2] : 0.0f;
            y[rbase + r2] = acc[r2] + b;
        }
    }
}

// ---------------------------------------------------------------------------
// Small elementwise / reduction kernels
// ---------------------------------------------------------------------------
__global__ void embed_kernel(const int* __restrict__ input,
                             const float* __restrict__ emb,
                             float* __restrict__ x)
{
    const int i = blockIdx.x * blockDim.x + threadIdx.x;
    if (i < E) x[i] = emb[(size_t)input[0] * (size_t)E + i];
}

__device__ __forceinline__ float sigmoid_f(float v) {
    return 1.0f / (1.0f + __expf(-v));
}

__global__ void gru_combine_kernel(const float* __restrict__ gi,
                                   const float* __restrict__ gh,
                                   const float* __restrict__ hprev,
                                   float* __restrict__ hout_ws,
                                   float* __restrict__ hout_final)
{
    const int i = blockIdx.x * blockDim.x + threadIdx.x;
    if (i < H) {
        const float r = sigmoid_f(gi[i]         + gh[i]);
        const float z = sigmoid_f(gi[i + H]     + gh[i + H]);
        const float n = tanhf    (gi[i + 2 * H] + r * gh[i + 2 * H]);
        const float h = (1.0f - z) * n + z * hprev[i];
        hout_ws[i]    = h;
        hout_final[i] = h;
    }
}

// softmax over L=512 scores (single block of 512 threads); writes attn to ws
// (in-place over score) and to d_out's attn3 slot.
__global__ void softmax_kernel(float* __restrict__ score,
                               const int* __restrict__ mask,
                               float* __restrict__ attn_out)
{
    __shared__ float red[512];
    const int t = threadIdx.x;
    float s = score[t];
    if (mask[t] == 0) s = -1.0e10f;
    red[t] = s;
    __syncthreads();
    for (int off = 256; off > 0; off >>= 1) {
        if (t < off) red[t] = fmaxf(red[t], red[t + off]);
        __syncthreads();
    }
    const float mx = red[0];
    __syncthreads();
    const float e = __expf(s - mx);
    red[t] = e;
    __syncthreads();
    for (int off = 256; off > 0; off >>= 1) {
        if (t < off) red[t] += red[t + off];
        __syncthreads();
    }
    const float a = e / red[0];
    score[t]    = a;   // ws attn (in place)
    attn_out[t] = a;   // output attn3
}

// c_t[h] = sum_l attn[l] * enc[l*H + h]  (threads across h -> coalesced per l)
__global__ void ctx_kernel(const float* __restrict__ enc,
                           const float* __restrict__ attn,
                           float* __restrict__ ct)
{
    const int h = blockIdx.x * blockDim.x + threadIdx.x;
    if (h < H) {
        float acc = 0.0f;
        for (int l = 0; l < L; ++l) acc += attn[l] * enc[(size_t)l * H + h];
        ct[h] = acc;
    }
}

// p_g = sigmoid(x.Vg + h1.Ug + ct.Wg)  (single block of 1024 threads)
__global__ void pg_kernel(const float* __restrict__ x,
                          const float* __restrict__ h1,
                          const float* __restrict__ ct,
                          const float* __restrict__ Vg,
                          const float* __restrict__ Ug,
                          const float* __restrict__ Wg,
                          float* __restrict__ pg)
{
    __shared__ float red[1024];
    const int t = threadIdx.x;
    float s = 0.0f;
    if (t < E) s += x[t] * Vg[t];
    s += h1[t] * Ug[t] + ct[t] * Wg[t];
    red[t] = s;
    __syncthreads();
    for (int off = 512; off > 0; off >>= 1) {
        if (t < off) red[t] += red[t + off];
        __syncthreads();
    }
    if (t == 0) pg[0] = sigmoid_f(red[0]);
}

__global__ void tanh_combine_kernel(const float* __restrict__ u,
                                    const float* __restrict__ w,
                                    float* __restrict__ t_)
{
    const int i = blockIdx.x * blockDim.x + threadIdx.x;
    if (i < H) t_[i] = tanhf(u[i] + w[i]);
}

// logsumexp over v of p_g * gvoc[v]  (single block of 1024 threads)
__global__ void lse_kernel(const float* __restrict__ gvoc,
                           const float* __restrict__ pg,
                           float* __restrict__ lse)
{
    __shared__ float red[1024];
    const int t = threadIdx.x;
    const float p = pg[0];
    float m = -3.0e38f;
    for (int v = t; v < V; v += 1024) m = fmaxf(m, p * gvoc[v]);
    red[t] = m;
    __syncthreads();
    for (int off = 512; off > 0; off >>= 1) {
        if (t < off) red[t] = fmaxf(red[t], red[t + off]);
        __syncthreads();
    }
    const float mx = red[0];
    __syncthreads();
    float s = 0.0f;
    for (int v = t; v < V; v += 1024) s += __expf(p * gvoc[v] - mx);
    red[t] = s;
    __syncthreads();
    for (int off = 512; off > 0; off >>= 1) {
        if (t < off) red[t] += red[t + off];
        __syncthreads();
    }
    if (t == 0) lse[0] = mx + __logf(red[0]);
}

// g[l, v] = p_g*gvoc[v] - lse   (row-constant attn shift cancels in log_softmax,
// so all 512 rows are identical). float4-vectorized stores; 32000 % 4 == 0 and
// rows never straddle a float4.
__global__ void final_write_kernel(const float* __restrict__ gvoc,
                                   const float* __restrict__ pg,
                                   const float* __restrict__ lse,
                                   float* __restrict__ g)
{
    const int idx = blockIdx.x * blockDim.x + threadIdx.x;  // 0 .. 4,096,000-1
    const float p  = pg[0];
    const float ls = lse[0];
    const int v4 = (idx % (V / 4)) * 4;
    const float4 gv = *reinterpret_cast<const float4*>(&gvoc[v4]);
    float4 o;
    o.x = p * gv.x - ls;
    o.y = p * gv.y - ls;
    o.z = p * gv.z - ls;
    o.w = p * gv.w - ls;
    *reinterpret_cast<float4*>(&g[(size_t)idx * 4]) = o;
}

// ---------------------------------------------------------------------------
// Launch
// ---------------------------------------------------------------------------
extern "C" void kernel_launch(void* const* d_in, const int* in_sizes, int n_in,
                              void* d_out, int out_size, void* d_ws, size_t ws_size,
                              hipStream_t stream)
{
    (void)in_sizes; (void)n_in; (void)out_size; (void)ws_size;

    const int*   input  = (const int*)  d_in[0];
    const float* hidden = (const float*)d_in[1];   // (2,1,H)
    const float* enc    = (const float*)d_in[2];   // (L,1,H)
    const int*   mask   = (const int*)  d_in[3];   // (1,L)
    const float* emb    = (const float*)d_in[5];   // (V,E)
    const float* W_ih0  = (const float*)d_in[6];   // (3H,E)
    const float* W_hh0  = (const float*)d_in[7];   // (3H,H)
    const float* b_ih0  = (const float*)d_in[8];
    const float* b_hh0  = (const float*)d_in[9];
    const float* W_ih1  = (const float*)d_in[10];  // (3H,H)
    const float* W_hh1  = (const float*)d_in[11];  // (3H,H)
    const float* b_ih1  = (const float*)d_in[12];
    const float* b_hh1  = (const float*)d_in[13];
    const float* Wh_W   = (const float*)d_in[14];  // (H,H)
    const float* Uh_W   = (const float*)d_in[15];  // (H,H)
    const float* out_W  = (const float*)d_in[16];  // (V,H)
    const float* out_b  = (const float*)d_in[17];  // (V,)
    const float* Wg_W   = (const float*)d_in[18];  // (1,H)
    const float* Ug_W   = (const float*)d_in[19];  // (1,H)
    const float* Vg_W   = (const float*)d_in[20];  // (1,E)

    // Output layout: g (L*V) | new_hidden (2*H) | attn3 (L)
    float* out      = (float*)d_out;
    float* g_out    = out;
    float* hid_out  = out + (size_t)L * (size_t)V;
    float* attn_out = hid_out + 2 * H;

    // Workspace (~206 KB of floats)
    float* ws    = (float*)d_ws;
    float* x     = ws;             // E
    float* gi0   = x    + E;       // 3H
    float* gh0   = gi0  + 3 * H;   // 3H
    float* h0    = gh0  + 3 * H;   // H
    float* gi1   = h0   + H;       // 3H
    float* gh1   = gi1  + 3 * H;   // 3H
    float* h1    = gh1  + 3 * H;   // H
    float* attn  = h1   + H;       // L (score, then attn in place)
    float* ct    = attn + L;       // H
    float* pg    = ct   + H;       // 1
    float* uvec  = pg   + 1;       // H
    float* wvec  = uvec + H;       // H
    float* tvec  = wvec + H;       // H
    float* gvoc  = tvec + H;       // V
    float* lse   = gvoc + V;       // 1

    // 1) embedding
    embed_kernel<<<2, 256, 0, stream>>>(input, emb, x);
    // 2) GRU layer 0
    gemv_wmma_kernel<<<48, 128, 0, stream>>>(W_ih0, x,      b_ih0, gi0, E);
    gemv_wmma_kernel<<<48, 128, 0, stream>>>(W_hh0, hidden, b_hh0, gh0, H);
    gru_combine_kernel<<<4, 256, 0, stream>>>(gi0, gh0, hidden, h0, hid_out);
    // 3) GRU layer 1
    gemv_wmma_kernel<<<48, 128, 0, stream>>>(W_ih1, h0,         b_ih1, gi1, H);
    gemv_wmma_kernel<<<48, 128, 0, stream>>>(W_hh1, hidden + H, b_hh1, gh1, H);
    gru_combine_kernel<<<4, 256, 0, stream>>>(gi1, gh1, hidden + H, h1, hid_out + H);
    // 4) attention scores + softmax + context
    gemv_wmma_kernel<<<8, 128, 0, stream>>>(enc, h1, nullptr, attn, H);
    softmax_kernel<<<1, 512, 0, stream>>>(attn, mask, attn_out);
    ctx_kernel<<<4, 256, 0, stream>>>(enc, attn, ct);
    // 5) pointer gate
    pg_kernel<<<1, 1024, 0, stream>>>(x, h1, ct, Vg_W, Ug_W, Wg_W, pg);
    // 6) vocab generator: t = tanh(Uh@h1 + Wh@ct); gvoc = out_W@t + out_b
    gemv_wmma_kernel<<<16, 128, 0, stream>>>(Uh_W, h1, nullptr, uvec, H);
    gemv_wmma_kernel<<<16, 128, 0, stream>>>(Wh_W, ct, nullptr, wvec, H);
    tanh_combine_kernel<<<4, 256, 0, stream>>>(uvec, wvec, tvec);
    gemv_wmma_kernel<<<500, 128, 0, stream>>>(out_W, tvec, out_b, gvoc, H);
    // 7) log-softmax (rows identical after the row-constant shift cancels)
    lse_kernel<<<1, 1024, 0, stream>>>(gvoc, pg, lse);
    final_write_kernel<<<(L * V / 4) / 256, 256, 0, stream>>>(gvoc, pg, lse, g_out);
}